// UpsampleInterp_29291676959107
// MI455X (gfx1250) — compile-verified
//
#include <hip/hip_runtime.h>
#include <hip/hip_bf16.h>

typedef float v4f __attribute__((ext_vector_type(4)));
typedef int   v4i __attribute__((ext_vector_type(4)));

// ---- geometry -------------------------------------------------------------
#define IN_D 80
#define IN_H 96
#define IN_W 112
#define CH   16
#define OUT_D 160
#define OUT_H 192
#define OUT_W 224

// output tile per workgroup
#define TD_OUT 8
#define TH_OUT 8
#define TW_OUT 32
// input tile (+1 clamped halo)
#define TD_IN 5
#define TH_IN 5
#define TW_IN 17
#define TILE_VOX   (TD_IN * TH_IN * TW_IN)     // 425
#define TILE_V4    (TILE_VOX * 4)              // 1700 float4 transfers
#define TILE_FLOATS (TILE_VOX * CH)            // 6800 floats = 27200 B LDS

// output strides in floats
#define OSTR_W  (CH)                 // 16
#define OSTR_H  (OUT_W * CH)         // 3584
#define OSTR_D  (OUT_H * OUT_W * CH) // 688128

// ---- CDNA5 async global->LDS copy (16B per lane) --------------------------
__device__ __forceinline__ void async_copy16(const float* __restrict__ g, float* l) {
#if defined(__gfx1250__)
# if __has_builtin(__builtin_amdgcn_global_load_async_to_lds_b128)
  __builtin_amdgcn_global_load_async_to_lds_b128(
      (__attribute__((address_space(1))) v4i*)g,
      (__attribute__((address_space(3))) v4i*)l,
      0, 0);
# else
  unsigned lds_addr =
      (unsigned)(unsigned long long)(__attribute__((address_space(3))) float*)l;
  asm volatile("global_load_async_to_lds_b128 %0, %1, off"
               :: "v"(lds_addr), "v"(g)
               : "memory");
# endif
#else
  *(v4f*)l = *(const v4f*)g;   // host-pass / non-gfx1250 placeholder
#endif
}

__device__ __forceinline__ void wait_async0() {
#if defined(__gfx1250__)
# if __has_builtin(__builtin_amdgcn_s_wait_asynccnt)
  __builtin_amdgcn_s_wait_asynccnt(0);
# else
  asm volatile("s_wait_asynccnt 0x0" ::: "memory");
# endif
#endif
}

__device__ __forceinline__ void store_nt16(float* p, v4f v) {
  __builtin_nontemporal_store(v, (v4f*)p);
}

// ---- kernel ---------------------------------------------------------------
__global__ void __launch_bounds__(256)
upsample2x_trilinear_kernel(const float* __restrict__ V, float* __restrict__ out) {
  __shared__ __align__(16) float lds[TILE_FLOATS];

  const int tileW = blockIdx.x;   // 0..6
  const int tileH = blockIdx.y;   // 0..23
  const int tileD = blockIdx.z;   // 0..19

  const int idb = tileD * (TD_OUT / 2);   // input base d
  const int ihb = tileH * (TH_OUT / 2);   // input base h
  const int iwb = tileW * (TW_OUT / 2);   // input base w

  // ---- stage input tile (with clamped +1 halo) into LDS, async B128 ------
  for (int e = threadIdx.x; e < TILE_V4; e += 256) {
    const int c4  = e & 3;
    const int vox = e >> 2;                 // 0..424
    const int tx  = vox % TW_IN;
    const int zy  = vox / TW_IN;
    const int ty  = zy % TH_IN;
    const int tz  = zy / TH_IN;
    const int gz = min(idb + tz, IN_D - 1);
    const int gy = min(ihb + ty, IN_H - 1);
    const int gx = min(iwb + tx, IN_W - 1);
    const float* src =
        V + ((((size_t)gz * IN_H + gy) * IN_W + gx) * CH + c4 * 4);
    // LDS layout: flat ((tz*5+ty)*17+tx)*16 + c4*4  ==  e*4 floats
    async_copy16(src, &lds[(size_t)e * 4]);
  }
  wait_async0();
  __syncthreads();

  // ---- compute: each thread owns (c4, owl) column, four of eight ohl rows --
  const int t    = threadIdx.x;
  const int c4   = t & 3;
  const int owl  = (t >> 2) & 31;   // 0..31
  const int ohl0 = (t >> 7);        // 0 or 1

  const int iw0 = owl >> 1;
  const int iw1 = iw0 + (owl & 1);  // duplicated index when even (exact)

  const v4f* L = (const v4f*)lds;

  float* outBase = out +
      (((size_t)tileD * TD_OUT) * (size_t)OUT_H +
        (size_t)tileH * TH_OUT) * (size_t)OUT_W * CH +
      (size_t)(tileW * TW_OUT + owl) * CH + (size_t)c4 * 4;

  for (int ohl = ohl0; ohl < TH_OUT; ohl += 2) {
    const int ih0 = ohl >> 1;
    const int ih1 = ih0 + (ohl & 1);

    auto hw = [&](int id) -> v4f {
      const int b0 = (id * TH_IN + ih0) * TW_IN;
      const int b1 = (id * TH_IN + ih1) * TW_IN;
      v4f a = L[(b0 + iw0) * 4 + c4];
      v4f b = L[(b0 + iw1) * 4 + c4];
      v4f c = L[(b1 + iw0) * 4 + c4];
      v4f d = L[(b1 + iw1) * 4 + c4];
      return 0.5f * (0.5f * (a + b) + 0.5f * (c + d));
    };

    float* po = outBase + (size_t)ohl * OSTR_H;

    // sliding window over input depth: 5 hw-interps -> 8 outputs
    v4f Xp = hw(0);
    #pragma unroll
    for (int k = 1; k <= TD_OUT / 2; ++k) {
      v4f Xk = hw(k);
      store_nt16(po + (size_t)(2 * k - 2) * OSTR_D, Xp);
      store_nt16(po + (size_t)(2 * k - 1) * OSTR_D, 0.5f * (Xp + Xk));
      Xp = Xk;
    }
  }
}

// ---- launch ---------------------------------------------------------------
extern "C" void kernel_launch(void* const* d_in, const int* in_sizes, int n_in,
                              void* d_out, int out_size, void* d_ws, size_t ws_size,
                              hipStream_t stream) {
  (void)in_sizes; (void)n_in; (void)d_ws; (void)ws_size; (void)out_size;
  const float* V = (const float*)d_in[0];
  float* out = (float*)d_out;
  dim3 grid(OUT_W / TW_OUT, OUT_H / TH_OUT, OUT_D / TD_OUT);  // 7 x 24 x 20
  upsample2x_trilinear_kernel<<<grid, dim3(256), 0, stream>>>(V, out);
}